// Q_net_3736621548252
// MI455X (gfx1250) — compile-verified
//
#include <hip/hip_runtime.h>
#include <math.h>

typedef float v2f __attribute__((ext_vector_type(2)));
typedef float v8f __attribute__((ext_vector_type(8)));

#define SQ2   1.41421356237309515f
#define IS3   0.57735026918962584f
#define IS5   0.44721359549995793f
#define IS6   0.40824829046386307f
#define IS8   0.35355339059327373f
#define S3F   1.73205080756887729f
#define S5F   2.23606797749978969f
#define S15F  3.87298334620741689f
#define A0F   0.28867513459481287f
#define A1F   0.5f
#define A2F   0.5f
#define R2C   0.70710678118654752f
#define R6C   0.40824829046386307f
#define PIF   3.14159265358979323846f
#define RMAXF 2.0f
#define RMINF 0.1f

#if __has_builtin(__builtin_amdgcn_wmma_f32_16x16x4_f32)
#define USE_WMMA 1
__device__ __forceinline__ v8f wmma4(v2f a, v2f b, v8f c) {
  // 8-arg pattern: (neg_a, A, neg_b, B, c_mod, C, reuse_a, reuse_b)
  return __builtin_amdgcn_wmma_f32_16x16x4_f32(false, a, false, b, (short)0, c, false, false);
}
#endif

// ---------------------------------------------------------------------------
// Constant coupling tensors C212 (5x3x5) and C222 (5x5x5), built on device.
// ---------------------------------------------------------------------------
__global__ void k_const(float* __restrict__ C212, float* __restrict__ C222) {
  if (threadIdx.x != 0 || blockIdx.x != 0) return;
  float M[5][3][3] = {};
  M[0][0][2] = M[0][2][0] = R2C;
  M[1][0][1] = M[1][1][0] = R2C;
  M[2][0][0] = -R6C; M[2][1][1] = 2.f * R6C; M[2][2][2] = -R6C;
  M[3][1][2] = M[3][2][1] = R2C;
  M[4][0][0] = -R2C; M[4][2][2] = R2C;
  float EPS[3][3][3] = {};
  EPS[0][1][2] = EPS[1][2][0] = EPS[2][0][1] = 1.f;
  EPS[0][2][1] = EPS[2][1][0] = EPS[1][0][2] = -1.f;

  float n1 = 0.f;
  for (int k = 0; k < 5; ++k)
    for (int i = 0; i < 3; ++i)
      for (int m = 0; m < 5; ++m) {
        float a = 0.f;
        for (int b = 0; b < 3; ++b)
          for (int c = 0; c < 3; ++c)
            for (int d = 0; d < 3; ++d)
              a += EPS[i][b][c] * M[k][b][d] * M[m][c][d];
        C212[k * 15 + i * 5 + m] = a;
        n1 += a * a;
      }
  n1 = 1.f / sqrtf(n1);
  for (int i = 0; i < 75; ++i) C212[i] *= n1;

  float n2 = 0.f;
  for (int k = 0; k < 5; ++k)
    for (int l = 0; l < 5; ++l)
      for (int m = 0; m < 5; ++m) {
        float a = 0.f;
        for (int x = 0; x < 3; ++x)
          for (int b = 0; b < 3; ++b)
            for (int c = 0; c < 3; ++c)
              a += M[k][x][b] * M[l][b][c] * M[m][c][x];
        C222[k * 25 + l * 5 + m] = a;
        n2 += a * a;
      }
  n2 = 1.f / sqrtf(n2);
  for (int i = 0; i < 125; ++i) C222[i] *= n2;
}

// ---------------------------------------------------------------------------
// Zero node accumulators and output.
// ---------------------------------------------------------------------------
__global__ void k_zero(float* __restrict__ node1, float* __restrict__ node2,
                       float* __restrict__ node3, float* __restrict__ out,
                       int n40, int n, int nf) {
  int i = blockIdx.x * blockDim.x + threadIdx.x;
  if (i < n40) { node1[i] = 0.f; node2[i] = 0.f; }
  if (i < n)   node3[i] = 0.f;
  if (i < nf)  out[i]   = 0.f;
}

// ---------------------------------------------------------------------------
// Small MLP: 1 -> 8 -> 8 -> 12   (VALU; 168 MACs, negligible)
// ---------------------------------------------------------------------------
__device__ __forceinline__ void fc12(float emb, const float* __restrict__ W0,
                                     const float* __restrict__ W1,
                                     const float* __restrict__ W2,
                                     float* __restrict__ o) {
  float h0[8], h1[8];
#pragma unroll
  for (int j = 0; j < 8; ++j) h0[j] = fmaxf(emb * W0[j], 0.f) * SQ2;
#pragma unroll
  for (int r = 0; r < 8; ++r) {
    float a = 0.f;
#pragma unroll
    for (int c = 0; c < 8; ++c) a += h0[c] * W1[c * 8 + r];
    h1[r] = fmaxf(a * IS8, 0.f) * SQ2;
  }
#pragma unroll
  for (int k = 0; k < 12; ++k) {
    float a = 0.f;
#pragma unroll
    for (int c = 0; c < 8; ++c) a += h1[c] * W2[c * 12 + k];
    o[k] = a * IS8;
  }
}

// ---------------------------------------------------------------------------
// Edge geometry + fc1 + layer-1 tensor product + scatter.
// ---------------------------------------------------------------------------
__global__ void k_edge_l1(const float* __restrict__ pos, const float* __restrict__ act,
                          const int* __restrict__ esrc, const int* __restrict__ edst,
                          const float* __restrict__ w10, const float* __restrict__ w11,
                          const float* __restrict__ w12,
                          float* __restrict__ embB, float* __restrict__ sh1B,
                          float* __restrict__ sh2B, float* __restrict__ node1,
                          int E, float inv) {
  int e = blockIdx.x * blockDim.x + threadIdx.x;
  if (e >= E) return;
  int s = esrc[e], d = edst[e];
  float ex = pos[s * 3 + 0] - pos[d * 3 + 0];
  float ey = pos[s * 3 + 1] - pos[d * 3 + 1];
  float ez = pos[s * 3 + 2] - pos[d * 3 + 2];
  float r2 = ex * ex + ey * ey + ez * ez;
  bool nz = r2 > 0.f;
  float rn = nz ? sqrtf(r2) : 0.f;
  float ir = nz ? 1.f / rn : 0.f;
  float x = ex * ir, y = ey * ir, z = ez * ir;
  float s1v[3] = { S3F * x, S3F * y, S3F * z };
  float s2v[5] = { S15F * x * z, S15F * x * y,
                   S5F * (y * y - 0.5f * (x * x + z * z)),
                   S15F * y * z, 0.5f * S15F * (z * z - x * x) };
  float crit = (rn < RMAXF && rn > RMINF) ? 1.f : 0.f;
  float emb = (cosf(rn * (PIF / RMAXF)) + 1.f) * 0.5f * crit + ((rn <= RMINF) ? 1.f : 0.f);
  embB[e] = emb;
#pragma unroll
  for (int i = 0; i < 3; ++i) sh1B[e * 3 + i] = s1v[i];
#pragma unroll
  for (int m = 0; m < 5; ++m) sh2B[e * 5 + m] = s2v[m];

  float w1o[12];
  fc12(emb, w10, w11, w12, w1o);

  float f0 = act[s * 3 + 0], f1 = act[s * 3 + 1], f2 = act[s * 3 + 2];
  float cr[3] = { f1 * s1v[2] - f2 * s1v[1],
                  f2 * s1v[0] - f0 * s1v[2],
                  f0 * s1v[1] - f1 * s1v[0] };
  float q00 = -s2v[2] * R6C - s2v[4] * R2C;
  float q11 = 2.f * s2v[2] * R6C;
  float q22 = -s2v[2] * R6C + s2v[4] * R2C;
  float q01 = s2v[1] * R2C, q02 = s2v[0] * R2C, q12 = s2v[3] * R2C;
  float qf[3] = { q00 * f0 + q01 * f1 + q02 * f2,
                  q01 * f0 + q11 * f1 + q12 * f2,
                  q02 * f0 + q12 * f1 + q22 * f2 };
  float fv[3] = { f0, f1, f2 };
  float* nd = node1 + (long)d * 40;
#pragma unroll
  for (int w = 0; w < 4; ++w)
#pragma unroll
    for (int i = 0; i < 3; ++i) {
      float o = w1o[w] * fv[i] * IS3 + w1o[4 + w] * cr[i] * IS6 + w1o[8 + w] * qf[i] * IS5;
      atomicAdd(nd + 8 + w * 3 + i, inv * o);
    }
}

// ---------------------------------------------------------------------------
// Layer 2: fc2 (1->8->8->224) on 32-edge tiles per wave via fp32 WMMA,
// then the full set of equivariant contractions, scatter to node2.
// A-fragments use clamped-index unconditional loads + multiplicative mask so
// no EXEC-toggling branches appear anywhere in the WMMA region.
// ---------------------------------------------------------------------------
__global__ void __launch_bounds__(256) k_layer2(
    const int* __restrict__ esrc, const int* __restrict__ edst,
    const float* __restrict__ embB, const float* __restrict__ sh1B,
    const float* __restrict__ sh2B, const float* __restrict__ node1,
    float* __restrict__ node2,
    const float* __restrict__ w20, const float* __restrict__ w21,
    const float* __restrict__ w22,
    const float* __restrict__ C212, const float* __restrict__ C222,
    int E, float inv) {
  __shared__ float tb[8][32][17];  // per-wave 32-edge x 16-ch tile, stride 17 = conflict-free
  const int lane = threadIdx.x & 31;
  const int wid = threadIdx.x >> 5;
  const int l15 = lane & 15;
  const bool hihalf = lane >= 16;

  int base = (blockIdx.x * 8 + wid) * 32;
  int eRaw = base + lane;
  bool valid = eRaw < E;
  int e = valid ? eRaw : (E - 1);  // clamp: keep EXEC all-ones through WMMA region
  int srcn = esrc[e], dstn = edst[e];

  const float* g1 = node1 + (long)srcn * 40;
  __builtin_prefetch(g1, 0, 1);

  float emb = embB[e];
  float sh1v[3], sh2v[5];
#pragma unroll
  for (int i = 0; i < 3; ++i) sh1v[i] = sh1B[e * 3 + i];
#pragma unroll
  for (int m = 0; m < 5; ++m) sh2v[m] = sh2B[e * 5 + m];

  // hidden 0 (per-lane, 1->8)
  float h0v[8];
#pragma unroll
  for (int j = 0; j < 8; ++j) h0v[j] = fmaxf(emb * w20[j], 0.f) * SQ2;

  // hidden 1 (8->8) : H1^T = (W21^T/sqrt8, 16x8-padded) x H0^T  per 16-edge group
  float h1v[8];
#if defined(USE_WMMA)
  {
    // A-fragments: depend only on ks, not on edge group -> hoisted.
    // Rows 8..15 of the padded W^T are zeroed by a multiplicative mask
    // (index clamped with &7 keeps every load in-bounds, no branches).
    const float amask = (l15 < 8) ? IS8 : 0.f;
    const int r8 = l15 & 7;
    v2f aF[2];
#pragma unroll
    for (int ks = 0; ks < 2; ++ks) {
      int cb = ks * 4 + (hihalf ? 2 : 0);
      aF[ks].x = w21[cb * 8 + r8] * amask;
      aF[ks].y = w21[(cb + 1) * 8 + r8] * amask;
    }
    v8f dg0 = {}, dg1 = {};
#pragma unroll
    for (int g = 0; g < 2; ++g) {
      v8f acc = {};
#pragma unroll
      for (int ks = 0; ks < 2; ++ks) {
        int nsrc = g * 16 + l15;
        float b0lo = __shfl(h0v[ks * 4 + 0], nsrc, 32);
        float b0hi = __shfl(h0v[ks * 4 + 2], nsrc, 32);
        float b1lo = __shfl(h0v[ks * 4 + 1], nsrc, 32);
        float b1hi = __shfl(h0v[ks * 4 + 3], nsrc, 32);
        v2f b;
        b.x = hihalf ? b0hi : b0lo;
        b.y = hihalf ? b1hi : b1lo;
        acc = wmma4(aF[ks], b, acc);
      }
      if (g == 0) dg0 = acc; else dg1 = acc;
    }
#pragma unroll
    for (int i = 0; i < 8; ++i) {
      float lo = __shfl(dg0[i], l15, 32);
      float hi = __shfl(dg1[i], l15, 32);
      h1v[i] = fmaxf(hihalf ? hi : lo, 0.f) * SQ2;
    }
  }
#else
#pragma unroll
  for (int r = 0; r < 8; ++r) {
    float a = 0.f;
#pragma unroll
    for (int c = 0; c < 8; ++c) a += h0v[c] * w21[c * 8 + r];
    h1v[r] = fmaxf(a * IS8, 0.f) * SQ2;
  }
#endif

  // gather node features (s[8], v[4][3], t[4][5])
  float sA[8], vA[12], tA[20];
#pragma unroll
  for (int u = 0; u < 8; ++u) sA[u] = g1[u];
#pragma unroll
  for (int k = 0; k < 12; ++k) vA[k] = g1[8 + k];
#pragma unroll
  for (int k = 0; k < 20; ++k) tA[k] = g1[20 + k];

  // per-edge geometric intermediates
  float q00 = -sh2v[2] * R6C - sh2v[4] * R2C;
  float q11 = 2.f * sh2v[2] * R6C;
  float q22 = -sh2v[2] * R6C + sh2v[4] * R2C;
  float q01 = sh2v[1] * R2C, q02 = sh2v[0] * R2C, q12 = sh2v[3] * R2C;
  float cr[12], qv[12];
#pragma unroll
  for (int u = 0; u < 4; ++u) {
    float vx = vA[u * 3], vy = vA[u * 3 + 1], vz = vA[u * 3 + 2];
    cr[u * 3 + 0] = vy * sh1v[2] - vz * sh1v[1];
    cr[u * 3 + 1] = vz * sh1v[0] - vx * sh1v[2];
    cr[u * 3 + 2] = vx * sh1v[1] - vy * sh1v[0];
    qv[u * 3 + 0] = q00 * vx + q01 * vy + q02 * vz;
    qv[u * 3 + 1] = q01 * vx + q11 * vy + q12 * vz;
    qv[u * 3 + 2] = q02 * vx + q12 * vy + q22 * vz;
  }
  float td[4];
#pragma unroll
  for (int u = 0; u < 4; ++u) {
    float a = 0.f;
#pragma unroll
    for (int m = 0; m < 5; ++m) a += tA[u * 5 + m] * sh2v[m];
    td[u] = a;
  }
  float c3[20] = {0.f}, c4[20] = {0.f};
#pragma unroll
  for (int m = 0; m < 5; ++m)
#pragma unroll
    for (int k = 0; k < 5; ++k) {
      float r3 = sh1v[0] * C212[k * 15 + m] + sh1v[1] * C212[k * 15 + 5 + m] +
                 sh1v[2] * C212[k * 15 + 10 + m];
      float r4 = 0.f;
#pragma unroll
      for (int l = 0; l < 5; ++l) r4 += sh2v[l] * C222[k * 25 + l * 5 + m];
#pragma unroll
      for (int u = 0; u < 4; ++u) {
        c3[u * 5 + m] += tA[u * 5 + k] * r3;
        c4[u * 5 + m] += tA[u * 5 + k] * r4;
      }
    }

  float out0[8] = {0.f}, out1[12] = {0.f}, out2[20] = {0.f};

  // fc2 output GEMM (8 -> 224), 16-channel tiles; WMMA -> LDS -> per-lane consume
#pragma unroll
  for (int t = 0; t < 14; ++t) {
#if defined(USE_WMMA)
    // A-fragment (W22^T slice / sqrt8): depends on (t, ks) only -> hoisted
    // out of the edge-group loop. All 16 rows are valid channels here.
    v2f aT[2];
#pragma unroll
    for (int ks = 0; ks < 2; ++ks) {
      int ch = t * 16 + l15;
      int cb = ks * 4 + (hihalf ? 2 : 0);
      aT[ks].x = w22[cb * 224 + ch] * IS8;
      aT[ks].y = w22[(cb + 1) * 224 + ch] * IS8;
    }
#pragma unroll
    for (int g = 0; g < 2; ++g) {
      v8f acc = {};
#pragma unroll
      for (int ks = 0; ks < 2; ++ks) {
        int nsrc = g * 16 + l15;
        float b0lo = __shfl(h1v[ks * 4 + 0], nsrc, 32);
        float b0hi = __shfl(h1v[ks * 4 + 2], nsrc, 32);
        float b1lo = __shfl(h1v[ks * 4 + 1], nsrc, 32);
        float b1hi = __shfl(h1v[ks * 4 + 3], nsrc, 32);
        v2f b;
        b.x = hihalf ? b0hi : b0lo;
        b.y = hihalf ? b1hi : b1lo;
        acc = wmma4(aT[ks], b, acc);
      }
      int el = g * 16 + l15;
      int chl = hihalf ? 8 : 0;
#pragma unroll
      for (int i = 0; i < 8; ++i) tb[wid][el][chl + i] = acc[i];
    }
#else
#pragma unroll
    for (int j = 0; j < 16; ++j) {
      int ch = t * 16 + j;
      float a = 0.f;
#pragma unroll
      for (int c = 0; c < 8; ++c) a += h1v[c] * w22[c * 224 + ch];
      tb[wid][lane][j] = a * IS8;
    }
#endif
    __syncthreads();
#pragma unroll
    for (int j = 0; j < 16; ++j) {
      int ch = t * 16 + j;             // compile-time after unroll
      float wv = tb[wid][lane][j];
      if (ch < 64) {
        int u = ch >> 3, w = ch & 7;
        out0[w] += (A0F * wv) * sA[u];
      } else if (ch < 96) {
        int u = (ch - 64) >> 2, w = (ch - 64) & 3;
        float tmp = (A2F * IS5) * wv * sA[u];
#pragma unroll
        for (int m = 0; m < 5; ++m) out2[w * 5 + m] += tmp * sh2v[m];
      } else if (ch < 112) {
        int u = (ch - 96) >> 2, w = (ch - 96) & 3;
        float ws = (A1F * IS3) * wv;
#pragma unroll
        for (int i = 0; i < 3; ++i) out1[w * 3 + i] += ws * vA[u * 3 + i];
      } else if (ch < 128) {
        int u = (ch - 112) >> 2, w = (ch - 112) & 3;
        float ws = (A1F * IS6) * wv;
#pragma unroll
        for (int i = 0; i < 3; ++i) out1[w * 3 + i] += ws * cr[u * 3 + i];
      } else if (ch < 144) {
        int u = (ch - 128) >> 2, w = (ch - 128) & 3;
        float ws = (A1F * IS5) * wv;
#pragma unroll
        for (int i = 0; i < 3; ++i) out1[w * 3 + i] += ws * qv[u * 3 + i];
      } else if (ch < 160) {
        int u = (ch - 144) >> 2, w = (ch - 144) & 3;
        float ws = (A2F * IS5) * wv;
#pragma unroll
        for (int m = 0; m < 5; ++m) out2[w * 5 + m] += ws * tA[u * 5 + m];
      } else if (ch < 176) {
        int u = (ch - 160) >> 2, w = (ch - 160) & 3;
        float ws = A2F * wv;
#pragma unroll
        for (int m = 0; m < 5; ++m) out2[w * 5 + m] += ws * c3[u * 5 + m];
      } else if (ch < 208) {
        int u = (ch - 176) >> 3, w = (ch - 176) & 7;
        out0[w] += (A0F * IS5 * wv) * td[u];
      } else {
        int u = (ch - 208) >> 2, w = (ch - 208) & 3;
        float ws = A2F * wv;
#pragma unroll
        for (int m = 0; m < 5; ++m) out2[w * 5 + m] += ws * c4[u * 5 + m];
      }
    }
    __syncthreads();
  }

  if (valid) {
    float* nd = node2 + (long)dstn * 40;
#pragma unroll
    for (int w = 0; w < 8; ++w) atomicAdd(nd + w, inv * out0[w]);
#pragma unroll
    for (int k = 0; k < 12; ++k) atomicAdd(nd + 8 + k, inv * out1[k]);
#pragma unroll
    for (int k = 0; k < 20; ++k) atomicAdd(nd + 20 + k, inv * out2[k]);
  }
}

// ---------------------------------------------------------------------------
// Layer 3: fc3 + scalar contraction + scatter.
// ---------------------------------------------------------------------------
__global__ void k_layer3(const int* __restrict__ esrc, const int* __restrict__ edst,
                         const float* __restrict__ embB, const float* __restrict__ sh2B,
                         const float* __restrict__ node2, float* __restrict__ node3,
                         const float* __restrict__ w30, const float* __restrict__ w31,
                         const float* __restrict__ w32, int E, float inv) {
  int e = blockIdx.x * blockDim.x + threadIdx.x;
  if (e >= E) return;
  int s = esrc[e], d = edst[e];
  float emb = embB[e];
  float sh2v[5];
#pragma unroll
  for (int m = 0; m < 5; ++m) sh2v[m] = sh2B[e * 5 + m];
  float w3o[12];
  fc12(emb, w30, w31, w32, w3o);
  const float* g = node2 + (long)s * 40;
  float acc = 0.f;
#pragma unroll
  for (int u = 0; u < 8; ++u) acc += g[u] * w3o[u];
  float acc2 = 0.f;
#pragma unroll
  for (int u = 0; u < 4; ++u) {
    float tdu = 0.f;
#pragma unroll
    for (int m = 0; m < 5; ++m) tdu += g[20 + u * 5 + m] * sh2v[m];
    acc2 += tdu * w3o[8 + u];
  }
  float eo = A0F * (acc + acc2 * IS5);
  atomicAdd(&node3[d], inv * eo);
}

// ---------------------------------------------------------------------------
// Per-frame reduction (NA = 256).
// ---------------------------------------------------------------------------
__global__ void k_reduce(const float* __restrict__ node3, float* __restrict__ out, int N) {
  int i = blockIdx.x * blockDim.x + threadIdx.x;
  if (i < N) atomicAdd(&out[i >> 8], node3[i]);
}

// ---------------------------------------------------------------------------
extern "C" void kernel_launch(void* const* d_in, const int* in_sizes, int n_in,
                              void* d_out, int out_size, void* d_ws, size_t ws_size,
                              hipStream_t stream) {
  const float* pos = (const float*)d_in[0];
  const float* act = (const float*)d_in[1];
  const int* esrc = (const int*)d_in[2];
  const int* edst = (const int*)d_in[3];
  const float* w10 = (const float*)d_in[4];
  const float* w11 = (const float*)d_in[5];
  const float* w12 = (const float*)d_in[6];
  const float* w20 = (const float*)d_in[7];
  const float* w21 = (const float*)d_in[8];
  const float* w22 = (const float*)d_in[9];
  const float* w30 = (const float*)d_in[10];
  const float* w31 = (const float*)d_in[11];
  const float* w32 = (const float*)d_in[12];

  int E = in_sizes[2];
  int N = in_sizes[0] / 3;
  int NF = N / 256;
  float inv = 1.0f / sqrtf((float)E / (float)N);

  float* wsf = (float*)d_ws;
  float* C212 = wsf;                       // 75  (pad to 80)
  float* C222 = wsf + 80;                  // 125 (pad to 128 -> 208)
  float* node1 = wsf + 208;                // N*40
  float* node2 = node1 + (size_t)N * 40;   // N*40
  float* node3 = node2 + (size_t)N * 40;   // N
  float* embB = node3 + N;                 // E
  float* sh1B = embB + (size_t)E;          // 3E
  float* sh2B = sh1B + (size_t)3 * E;      // 5E

  int n40 = N * 40;
  int eb = (E + 255) / 256;

  k_const<<<1, 1, 0, stream>>>(C212, C222);
  k_zero<<<(n40 + 255) / 256, 256, 0, stream>>>(node1, node2, node3, (float*)d_out, n40, N, NF);
  k_edge_l1<<<eb, 256, 0, stream>>>(pos, act, esrc, edst, w10, w11, w12,
                                    embB, sh1B, sh2B, node1, E, inv);
  k_layer2<<<eb, 256, 0, stream>>>(esrc, edst, embB, sh1B, sh2B, node1, node2,
                                   w20, w21, w22, C212, C222, E, inv);
  k_layer3<<<eb, 256, 0, stream>>>(esrc, edst, embB, sh2B, node2, node3,
                                   w30, w31, w32, E, inv);
  k_reduce<<<(N + 255) / 256, 256, 0, stream>>>(node3, (float*)d_out, N);
}